// ProxyNCA_24326694765114
// MI455X (gfx1250) — compile-verified
//
#include <hip/hip_runtime.h>
#include <math.h>

typedef __attribute__((ext_vector_type(2))) float v2f;
typedef __attribute__((ext_vector_type(8))) float v8f;

#define NB_CLASSES 100000
#define SZ_EMBED   64
#define SZ_BATCH   1024
#define ROWTILES   (SZ_BATCH / 16)   // 64 tiles of 16 rows
#define NCHUNKS    63                // 63 * 1600 = 100800 >= 100000
#define CPC        1600              // classes per chunk (multiple of 64)
#define K3_WAVES   4
#define K3_THREADS (K3_WAVES * 32)
#define ITERS      (CPC / (K3_WAVES * 16))  // 25 class-tiles per wave per chunk

__device__ __forceinline__ void lse_combine(float& m, float& s, float om, float os) {
    // merge two (max, sum-of-exp) partials; (m=-inf, s=0) is the identity
    if (os > 0.0f) {
        if (om > m) { s = s * __expf(m - om) + os; m = om; }
        else        { s = s + os * __expf(om - m); }
    }
}

// -------------------------------------------- K1: proxy norms -> (inv_norm, p_sq)
__global__ __launch_bounds__(256)
void k1_norms(const float* __restrict__ proxies, float2* __restrict__ normdata) {
    int c = blockIdx.x * 256 + threadIdx.x;
    if (c >= NB_CLASSES) return;
    const float4* pr = (const float4*)(proxies + (size_t)c * SZ_EMBED);
    float sq = 0.0f;
#pragma unroll
    for (int t = 0; t < SZ_EMBED / 4; ++t) {
        float4 v = pr[t];
        sq += v.x * v.x + v.y * v.y + v.z * v.z + v.w * v.w;
    }
    float inv = 1.0f / fmaxf(sqrtf(sq), 1e-12f);
    normdata[c] = make_float2(inv, sq * inv * inv);  // (1/||p||, ||P||^2 post-norm)
}

// ------------------------------------------------- K2: x_sq + positive distance
__global__ __launch_bounds__(256)
void k2_pos(const float* __restrict__ xs, const int* __restrict__ ys,
            const float* __restrict__ proxies, const float2* __restrict__ normdata,
            float* __restrict__ x_sq, float* __restrict__ d_pos) {
    int i = blockIdx.x * 256 + threadIdx.x;
    if (i >= SZ_BATCH) return;
    int y = ys[i];
    float2 nd = normdata[y];
    const float4* xr = (const float4*)(xs + (size_t)i * SZ_EMBED);
    const float4* pr = (const float4*)(proxies + (size_t)y * SZ_EMBED);
    float dot = 0.0f, xq = 0.0f;
#pragma unroll
    for (int t = 0; t < SZ_EMBED / 4; ++t) {
        float4 xv = xr[t], pv = pr[t];
        dot += xv.x * pv.x + xv.y * pv.y + xv.z * pv.z + xv.w * pv.w;
        xq  += xv.x * xv.x + xv.y * xv.y + xv.z * xv.z + xv.w * xv.w;
    }
    x_sq[i] = xq;
    float d = xq + nd.y - 2.0f * nd.x * dot;
    d_pos[i] = fmaxf(d, 0.0f);
}

// ------------- K3: fused f32 WMMA GEMM + online masked logsumexp over classes
__global__ __launch_bounds__(K3_THREADS)
void k3_gemm_lse(const float* __restrict__ xs, const int* __restrict__ ys,
                 const float* __restrict__ proxies,
                 const float2* __restrict__ normdata,
                 const float* __restrict__ x_sq,
                 float2* __restrict__ partials) {
    const int wave = threadIdx.x >> 5;
    const int lane = threadIdx.x & 31;
    const int l16  = lane & 15;
    const int hi   = lane >> 4;           // 0: rows 0..7 / K+0..1; 1: rows 8..15 / K+2..3
    const int row0 = blockIdx.x * 16;
    const int chunkBase = blockIdx.y * CPC;
    const float NEG_INF = -__builtin_inff();

    // Preload A fragments per ISA 16x4 f32 layout: lane l16 = row, hi half = K offset +2.
    v2f af[16];
    const float* arow = xs + (size_t)(row0 + l16) * SZ_EMBED + 2 * hi;
#pragma unroll
    for (int t = 0; t < 16; ++t)
        af[t] = *(const v2f*)(arow + 4 * t);

    // Per-lane row state: D-tile lane holds rows (hi*8 .. hi*8+7) for one class column.
    float xsq[8]; int myys[8]; float mrun[8], srun[8];
#pragma unroll
    for (int r = 0; r < 8; ++r) {
        int m = hi * 8 + r;
        xsq[r]  = x_sq[row0 + m];
        myys[r] = ys[row0 + m];
        mrun[r] = NEG_INF;
        srun[r] = 0.0f;
    }

    // Software-pipeline the per-class (inv_norm, p_sq) scalar one tile ahead so
    // its load never injects a wait into the WMMA chain.
    const int class0 = chunkBase + wave * 16 + l16;
    float2 ndcur = normdata[min(class0, NB_CLASSES - 1)];

    for (int it = 0; it < ITERS; ++it) {
        int myclass = class0 + it * (K3_WAVES * 16);
        int cclamp  = min(myclass, NB_CLASSES - 1);
        bool valid  = (myclass < NB_CLASSES);
        const float* prow = proxies + (size_t)cclamp * SZ_EMBED + 2 * hi;

        // Batch all 16 B-fragment loads (raw, unnormalized): one clause, staged waits.
        v2f bf[16];
#pragma unroll
        for (int t = 0; t < 16; ++t)
            bf[t] = *(const v2f*)(prow + 4 * t);

        // Prefetch next tile (WGP scope pulls into all cache levels) + next scalar.
        float2 ndnext = ndcur;
        if (it + 1 < ITERS) {
            int nc = min(myclass + K3_WAVES * 16, NB_CLASSES - 1);
            ndnext = normdata[nc];
            __builtin_prefetch(proxies + (size_t)nc * SZ_EMBED + 32 * hi, 0, 3);
        }

        v8f acc = {0.f, 0.f, 0.f, 0.f, 0.f, 0.f, 0.f, 0.f};
#pragma unroll
        for (int t = 0; t < 16; ++t)
            acc = __builtin_amdgcn_wmma_f32_16x16x4_f32(
                false, af[t], false, bf[t], (short)0, acc, false, false);

        // acc[r] = <x_row, p_raw_class>; normalization folded into the epilogue:
        // dist = x_sq + p_sq - 2 * inv_norm * acc.  Branchless masked update
        // (exclusion c == ys[row] is ~1-in-100k: selects beat exec-mask branches).
        float twoinv = 2.0f * ndcur.x;
#pragma unroll
        for (int r = 0; r < 8; ++r) {
            float dist  = fmaxf(xsq[r] + ndcur.y - twoinv * acc[r], 0.0f);
            float logit = -dist;
            bool  upd   = valid && (myclass != myys[r]);
            float mn    = fmaxf(mrun[r], logit);
            float sn    = srun[r] * __expf(mrun[r] - mn) + __expf(logit - mn);
            mrun[r] = upd ? mn : mrun[r];
            srun[r] = upd ? sn : srun[r];
        }
        ndcur = ndnext;
    }

    // Reduce across the 16 lanes sharing each row group (masks stay within halves)
#pragma unroll
    for (int mask = 1; mask <= 8; mask <<= 1) {
#pragma unroll
        for (int r = 0; r < 8; ++r) {
            float om = __shfl_xor(mrun[r], mask, 32);
            float os = __shfl_xor(srun[r], mask, 32);
            lse_combine(mrun[r], srun[r], om, os);
        }
    }

    // Cross-wave combine in LDS; wave w contributes 16 (m,s) pairs via lanes 0 and 16
    __shared__ float2 lds[K3_WAVES][16];
    if (l16 == 0) {
#pragma unroll
        for (int r = 0; r < 8; ++r)
            lds[wave][hi * 8 + r] = make_float2(mrun[r], srun[r]);
    }
    __syncthreads();
    if (threadIdx.x < 16) {
        float m = NEG_INF, s = 0.0f;
#pragma unroll
        for (int w = 0; w < K3_WAVES; ++w) {
            float2 p = lds[w][threadIdx.x];
            lse_combine(m, s, p.x, p.y);
        }
        partials[((size_t)blockIdx.y * ROWTILES + blockIdx.x) * 16 + threadIdx.x] =
            make_float2(m, s);
    }
}

// ------------------------------- K4: merge chunk partials, mean over the batch
__global__ __launch_bounds__(1024)
void k4_reduce(const float2* __restrict__ partials, const float* __restrict__ d_pos,
               float* __restrict__ out) {
    const float NEG_INF = -__builtin_inff();
    int i = threadIdx.x;                 // one thread per sample
    float m = NEG_INF, s = 0.0f;
    int rt = i >> 4, rr = i & 15;
    for (int c = 0; c < NCHUNKS; ++c) {
        float2 p = partials[((size_t)c * ROWTILES + rt) * 16 + rr];
        lse_combine(m, s, p.x, p.y);
    }
    float per = d_pos[i] + (m + logf(s));   // d_pos + log sum_neg exp(-dist)

    __shared__ float red[1024];
    red[i] = per;
    __syncthreads();
    for (int off = 512; off > 0; off >>= 1) {
        if (i < off) red[i] += red[i + off];
        __syncthreads();
    }
    if (i == 0) out[0] = red[0] / (float)SZ_BATCH;
}

// --------------------------------------------------------------------- launch
extern "C" void kernel_launch(void* const* d_in, const int* in_sizes, int n_in,
                              void* d_out, int out_size, void* d_ws, size_t ws_size,
                              hipStream_t stream) {
    const float* xs      = (const float*)d_in[0];
    const int*   ys      = (const int*)d_in[1];
    const float* proxies = (const float*)d_in[2];
    float* out = (float*)d_out;

    // Workspace layout (floats), ~1.33 MB total
    float*  ws       = (float*)d_ws;
    float2* normdata = (float2*)ws;                          // 100000 float2
    float*  x_sq     = ws + 2 * NB_CLASSES;                  // 1024
    float*  d_pos    = ws + 2 * NB_CLASSES + SZ_BATCH;       // 1024
    float2* partials = (float2*)(ws + 2 * NB_CLASSES + 2 * SZ_BATCH); // 63*64*16 float2

    k1_norms<<<(NB_CLASSES + 255) / 256, 256, 0, stream>>>(proxies, normdata);
    k2_pos<<<(SZ_BATCH + 255) / 256, 256, 0, stream>>>(xs, ys, proxies, normdata,
                                                       x_sq, d_pos);
    dim3 g3(ROWTILES, NCHUNKS);
    k3_gemm_lse<<<g3, K3_THREADS, 0, stream>>>(xs, ys, proxies, normdata, x_sq,
                                               partials);
    k4_reduce<<<1, 1024, 0, stream>>>(partials, d_pos, out);
}